// MMDLoss_22256520527948
// MI455X (gfx1250) — compile-verified
//
#include <hip/hip_runtime.h>
#include <hip/hip_bf16.h>
#include <stdint.h>

// ---------------------------------------------------------------------------
// MMD loss, MI455X (gfx1250, wave32).
//   source: 4096x256 f32, target: 4096x256 f32  -> scalar f32 loss.
//   G = X X^T via V_WMMA_F32_16X16X4_F32 (full fp32 matrix path),
//   staging via global_load_async_to_lds_b128 (ASYNCcnt),
//   5-kernel Gaussian sum with ONE v_exp_f32 per element (repeated squaring).
// ---------------------------------------------------------------------------

typedef __attribute__((ext_vector_type(2))) float v2f;
typedef __attribute__((ext_vector_type(8))) float v8f;

#define NHALF   4096
#define MTOT    8192
#define KDIM    256
#define BT      128          // block tile (rows == cols)
#define KC      128          // K staging chunk
#define LS      132          // padded LDS row stride (floats): 132 % 64 == 4 -> conflict-free

#define WS_SUMSQ  0
#define WS_SCALE  1
#define WS_LOSS   2
#define WS_SQ     16                   // 8192 floats: ||x_i||^2
#define WS_COLSUM (WS_SQ + MTOT)       // 256 floats: column sums
#define WS_FLOATS (WS_COLSUM + KDIM)

__device__ __forceinline__ void async_load_b128(uint32_t lds_byte_off, const float* g) {
  // VDST = LDS byte offset (VGPR), VADDR = 64-bit global address (VGPR pair).
  asm volatile("global_load_async_to_lds_b128 %0, %1, off"
               :: "v"(lds_byte_off), "v"((unsigned long long)(size_t)g)
               : "memory");
}

__device__ __forceinline__ void wait_async0() {
#if __has_builtin(__builtin_amdgcn_s_wait_asynccnt)
  __builtin_amdgcn_s_wait_asynccnt(0);
#else
  asm volatile("s_wait_asynccnt 0" ::: "memory");
#endif
}

// ---------------------------------------------------------------- init ws ---
__global__ void mmd_init(float* __restrict__ ws) {
  for (int i = threadIdx.x; i < WS_FLOATS; i += blockDim.x) ws[i] = 0.0f;
}

// ------------------------------------------------- row stats (one pass) -----
// sq[i] = sum_k x[i][k]^2 ; sumSq = sum_i sq[i] ; colsum[k] = sum_i x[i][k]
__global__ __launch_bounds__(256) void mmd_row_stats(const float* __restrict__ src,
                                                     const float* __restrict__ tgt,
                                                     float* __restrict__ ws) {
  __shared__ float cs[KDIM];
  const int tid = threadIdx.x;
  cs[tid] = 0.0f;
  __syncthreads();

  const int w = tid >> 5, lane = tid & 31;
  const int row = blockIdx.x * 8 + w;                 // 1024 blocks * 8 waves = 8192 rows
  const float* p = (row < NHALF) ? src + (size_t)row * KDIM
                                 : tgt + (size_t)(row - NHALF) * KDIM;
  const float4 a = *(const float4*)(p + lane * 8);
  const float4 b = *(const float4*)(p + lane * 8 + 4);

  float r = a.x*a.x + a.y*a.y + a.z*a.z + a.w*a.w
          + b.x*b.x + b.y*b.y + b.z*b.z + b.w*b.w;
  for (int o = 16; o > 0; o >>= 1) r += __shfl_xor(r, o, 32);   // wave32 reduce
  if (lane == 0) {
    ws[WS_SQ + row] = r;
    atomicAdd(&ws[WS_SUMSQ], r);
  }

  const int c0 = lane * 8;
  atomicAdd(&cs[c0 + 0], a.x); atomicAdd(&cs[c0 + 1], a.y);
  atomicAdd(&cs[c0 + 2], a.z); atomicAdd(&cs[c0 + 3], a.w);
  atomicAdd(&cs[c0 + 4], b.x); atomicAdd(&cs[c0 + 5], b.y);
  atomicAdd(&cs[c0 + 6], b.z); atomicAdd(&cs[c0 + 7], b.w);
  __syncthreads();
  atomicAdd(&ws[WS_COLSUM + tid], cs[tid]);
}

// ---------------------------------------------------------- bandwidth -------
// sum(d2) = 2*m*sumSq - 2*||colsum||^2 ; bw = sum(d2)/(m^2-m)/4
// scale = -log2(e) / (16*bw)  so  e = exp2(d2*scale) = exp(-d2/(16*bw))
__global__ __launch_bounds__(256) void mmd_finalize_bw(float* __restrict__ ws) {
  __shared__ float red[8];
  const int tid = threadIdx.x;
  const float v = ws[WS_COLSUM + tid];
  float p = v * v;
  for (int o = 16; o > 0; o >>= 1) p += __shfl_xor(p, o, 32);
  if ((tid & 31) == 0) red[tid >> 5] = p;
  __syncthreads();
  if (tid == 0) {
    float ss = 0.0f;
    for (int i = 0; i < 8; ++i) ss += red[i];
    const float m = (float)MTOT;
    const float sum_d2 = 2.0f * m * ws[WS_SUMSQ] - 2.0f * ss;
    float bw = sum_d2 / (m * m - m);
    bw *= 0.25f;                                   // / KERNEL_MUL^(KERNEL_NUM//2)
    ws[WS_SCALE] = -1.4426950408889634f / (16.0f * bw);
  }
}

// ------------------------------------------------------- main tile GEMM -----
__global__ __launch_bounds__(256) void mmd_tile(const float* __restrict__ src,
                                                const float* __restrict__ tgt,
                                                float* __restrict__ ws) {
  const int br = blockIdx.y;
  const int bc = blockIdx.x;
  if (bc < br) return;                              // symmetric: upper triangle only

  __shared__ float shA[BT * LS];
  __shared__ float shB[BT * LS];

  const float* pa = (br < 32) ? src + (size_t)br * BT * KDIM
                              : tgt + (size_t)(br - 32) * BT * KDIM;
  const float* pb = (bc < 32) ? src + (size_t)bc * BT * KDIM
                              : tgt + (size_t)(bc - 32) * BT * KDIM;

  const int tid  = threadIdx.x;
  const int w    = tid >> 5, lane = tid & 31;
  const int wr   = w >> 2,   wc   = w & 3;          // 2x4 wave grid -> 64x32 per wave
  const int hi   = lane >> 4, lo  = lane & 15;

  v8f acc[4][2] = {};                               // 8 x (16x16 f32) accumulators

  const uint32_t ldsA = (uint32_t)(size_t)shA;
  const uint32_t ldsB = (uint32_t)(size_t)shB;

  for (int kc = 0; kc < KDIM; kc += KC) {
    // Async stage 128 rows x 128 floats (A and B) : 16B per lane per issue.
    for (int idx = tid; idx < BT * (KC / 4); idx += 256) {
      const int row = idx >> 5;
      const int q   = idx & 31;
      const uint32_t loff = (uint32_t)(row * LS + q * 4) * 4u;
      async_load_b128(ldsA + loff, pa + row * KDIM + kc + q * 4);
      async_load_b128(ldsB + loff, pb + row * KDIM + kc + q * 4);
    }
    wait_async0();
    __syncthreads();

    // A-frag 16x4: lane(lo)=row M, hi selects K pair {k,k+1} / {k+2,k+3}.
    // B-frag 4x16: B[k][n] = Xb[n][k] -> identical gather from row-major LDS.
    const float* baseA = shA + (wr * 64 + lo) * LS + 2 * hi;
    const float* baseB = shB + (wc * 32 + lo) * LS + 2 * hi;
#pragma unroll 4
    for (int k = 0; k < KC; k += 4) {
      const v2f a0 = *(const v2f*)(baseA + 0 * 16 * LS + k);
      const v2f a1 = *(const v2f*)(baseA + 1 * 16 * LS + k);
      const v2f a2 = *(const v2f*)(baseA + 2 * 16 * LS + k);
      const v2f a3 = *(const v2f*)(baseA + 3 * 16 * LS + k);
      const v2f b0 = *(const v2f*)(baseB + 0 * 16 * LS + k);
      const v2f b1 = *(const v2f*)(baseB + 1 * 16 * LS + k);
      acc[0][0] = __builtin_amdgcn_wmma_f32_16x16x4_f32(false, a0, false, b0, (short)0, acc[0][0], false, false);
      acc[0][1] = __builtin_amdgcn_wmma_f32_16x16x4_f32(false, a0, false, b1, (short)0, acc[0][1], false, false);
      acc[1][0] = __builtin_amdgcn_wmma_f32_16x16x4_f32(false, a1, false, b0, (short)0, acc[1][0], false, false);
      acc[1][1] = __builtin_amdgcn_wmma_f32_16x16x4_f32(false, a1, false, b1, (short)0, acc[1][1], false, false);
      acc[2][0] = __builtin_amdgcn_wmma_f32_16x16x4_f32(false, a2, false, b0, (short)0, acc[2][0], false, false);
      acc[2][1] = __builtin_amdgcn_wmma_f32_16x16x4_f32(false, a2, false, b1, (short)0, acc[2][1], false, false);
      acc[3][0] = __builtin_amdgcn_wmma_f32_16x16x4_f32(false, a3, false, b0, (short)0, acc[3][0], false, false);
      acc[3][1] = __builtin_amdgcn_wmma_f32_16x16x4_f32(false, a3, false, b1, (short)0, acc[3][1], false, false);
    }
    __syncthreads();
  }

  // Epilogue: d2 = sq_i + sq_j - 2G ; ksum = e + e^2 + e^4 + e^8 + e^16,
  // e = exp2(d2*scale). One v_exp_f32 per element.
  const float scale = ws[WS_SCALE];
  const float* sq = ws + WS_SQ;
  const int ibase = br * BT + wr * 64;
  const int jbase = bc * BT + wc * 32;
  float lsum = 0.0f;
#pragma unroll
  for (int mt = 0; mt < 4; ++mt) {
#pragma unroll
    for (int nt = 0; nt < 2; ++nt) {
      const int j = jbase + nt * 16 + lo;
      const float sqj = sq[j];
#pragma unroll
      for (int v = 0; v < 8; ++v) {
        const int i = ibase + mt * 16 + v + 8 * hi;    // C/D layout: VGPR v -> M=v(+8)
        const float g  = acc[mt][nt][v];
        const float d2 = sq[i] + sqj - 2.0f * g;
        const float e  = __builtin_amdgcn_exp2f(d2 * scale);
        const float e2 = e * e, e4 = e2 * e2, e8 = e4 * e4, e16 = e8 * e8;
        lsum += e + e2 + e4 + e8 + e16;
      }
    }
  }
  for (int o = 16; o > 0; o >>= 1) lsum += __shfl_xor(lsum, o, 32);

  float wgt = ((br < 32) == (bc < 32)) ? 1.0f : -1.0f; // xx/yy vs xy/yx blocks
  if (bc > br) wgt *= 2.0f;                            // strict upper counted twice
  if (lane == 0) atomicAdd(&ws[WS_LOSS], wgt * lsum);
}

// ------------------------------------------------------------- finalize -----
__global__ void mmd_finalize_loss(const float* __restrict__ ws, float* __restrict__ out) {
  out[0] = ws[WS_LOSS] * (1.0f / ((float)NHALF * (float)NHALF));
}

// ---------------------------------------------------------------------------
extern "C" void kernel_launch(void* const* d_in, const int* in_sizes, int n_in,
                              void* d_out, int out_size, void* d_ws, size_t ws_size,
                              hipStream_t stream) {
  const float* src = (const float*)d_in[0];
  const float* tgt = (const float*)d_in[1];
  float* ws  = (float*)d_ws;
  float* out = (float*)d_out;

  mmd_init<<<1, 256, 0, stream>>>(ws);
  mmd_row_stats<<<MTOT / 8, 256, 0, stream>>>(src, tgt, ws);
  mmd_finalize_bw<<<1, 256, 0, stream>>>(ws);
  dim3 grid(MTOT / BT, MTOT / BT);                    // 64x64, lower triangle early-outs
  mmd_tile<<<grid, 256, 0, stream>>>(src, tgt, ws);
  mmd_finalize_loss<<<1, 1, 0, stream>>>(ws, out);
}